// FFTUnitary_46024869544016
// MI455X (gfx1250) — compile-verified
//
#include <hip/hip_runtime.h>

// FFT-style unitary: 10 butterfly stages + diagonal psi rotation over rows of
// N x 2D (D=1024 complex).  Reformulated as block-diagonal dense matmuls:
//   stages 0-3  -> 64  distinct 16x16 complex mats on contiguous 16-blocks
//   stages 4-7  -> 64  distinct 16x16 complex mats on stride-16 slices
//   stages 8-9  -> 256 distinct 4x4 complex mats on stride-256 slices (psi folded)
// Main kernel: 16 rows per workgroup resident in LDS (filled via CDNA5 async
// global->LDS loads, ASYNCcnt), stages A/B via V_WMMA_F32_16X16X4_F32
// (4 real GEMMs per complex tile), stage C via VALU.

#define DD    1024
#define PP    10
#define NN    16384
#define LDST  1028   // padded LDS row stride (floats): bank-conflict free, 16B aligned

typedef __attribute__((ext_vector_type(2))) float v2f;
typedef __attribute__((ext_vector_type(8))) float v8f;

// Per-stage 2x2 rotation coefficients for global position i at stage j.
// k = i with bit j removed; b = bit j of i selects even/odd interleave slot.
__device__ __forceinline__ void stage_coef(const float* __restrict__ theta,
                                           const float* __restrict__ phi,
                                           int j, int i,
                                           float& ur, float& ui, float& vr, float& vi) {
  int k = (i & ((1 << j) - 1)) | ((i >> (j + 1)) << j);
  int b = (i >> j) & 1;
  float th = theta[j * (DD / 2) + k];
  float ph = phi[j * (DD / 2) + k];
  float s1, c1, s2, c2;
  sincosf(th * 0.5f, &s1, &c1);
  sincosf(th * 0.5f + ph, &s2, &c2);
  if (b == 0) { ur = -s1 * s2; ui =  s1 * c2; vr = -c1 * s1; vi = c1 * c1; }
  else        { ur =  s1 * s1; ui = -c1 * s1; vr = -c1 * s2; vi = c1 * c2; }
}

// Build one 16x16 complex matrix (product of 4 butterfly stages) per block and
// store it pre-swizzled in WMMA A-operand order:
//   opRe/opIm[((g*4 + kc)*32 + lane)*2 + t] = M[lane&15][kc*4 + 2*(lane>>4) + t]
__global__ void build_mats16(const float* __restrict__ theta,
                             const float* __restrict__ phi,
                             float* __restrict__ opRe, float* __restrict__ opIm,
                             int stage0, int modeB) {
  __shared__ float Mre[256], Mim[256];
  const int g = blockIdx.x, tid = threadIdx.x;
  const int r = tid >> 4, c = tid & 15;
  Mre[tid] = (r == c) ? 1.0f : 0.0f;
  Mim[tid] = 0.0f;
  __syncthreads();
  // global index for row r of this block's matrix
  const int i = modeB ? ((g >> 4) * 256 + r * 16 + (g & 15)) : (g * 16 + r);
  for (int jj = 0; jj < 4; ++jj) {
    float ur, ui, vr, vi;
    stage_coef(theta, phi, stage0 + jj, i, ur, ui, vr, vi);
    const int p = ((r ^ (1 << jj)) << 4) | c;
    float mr = Mre[tid], mi = Mim[tid], pr = Mre[p], pi = Mim[p];
    __syncthreads();
    Mre[tid] = ur * mr - ui * mi + vr * pr - vi * pi;
    Mim[tid] = ur * mi + ui * mr + vr * pi + vi * pr;
    __syncthreads();
  }
  // swizzle into WMMA A-operand lane layout (ISA 7.12.2: 32-bit A 16x4)
  const int kc = tid >> 6, lane = (tid >> 1) & 31, t = tid & 1;
  const int M = lane & 15, K = kc * 4 + 2 * (lane >> 4) + t;
  const int o = ((g * 4 + kc) * 32 + lane) * 2 + t;
  opRe[o] = Mre[M * 16 + K];
  opIm[o] = Mim[M * 16 + K];
}

// Build 256 4x4 complex matrices (stages 8,9) with psi rotation folded in.
// CC[l*32 + (r*4+c)*2 + {0,1}] = (re,im) of C'_l[r][c]
__global__ void build_mats4(const float* __restrict__ theta,
                            const float* __restrict__ phi,
                            const float* __restrict__ psi,
                            float* __restrict__ CC) {
  __shared__ float Mre[256], Mim[256];
  const int tid = threadIdx.x;
  const int sub = tid >> 4;                 // which l within block
  const int l = blockIdx.x * 16 + sub;      // low 8 bits of global index
  const int e = tid & 15;
  const int r = e >> 2, c = e & 3;
  Mre[tid] = (r == c) ? 1.0f : 0.0f;
  Mim[tid] = 0.0f;
  __syncthreads();
  const int i = r * 256 + l;
  for (int jj = 0; jj < 2; ++jj) {
    float ur, ui, vr, vi;
    stage_coef(theta, phi, 8 + jj, i, ur, ui, vr, vi);
    const int p = (sub << 4) | (((r ^ (1 << jj)) << 2) | c);
    float mr = Mre[tid], mi = Mim[tid], pr = Mre[p], pi = Mim[p];
    __syncthreads();
    Mre[tid] = ur * mr - ui * mi + vr * pr - vi * pi;
    Mim[tid] = ur * mi + ui * mr + vr * pi + vi * pr;
    __syncthreads();
  }
  float sp, cp;
  sincosf(psi[i], &sp, &cp);                // row phase e^{i psi[r*256+l]}
  float mr = Mre[tid], mi = Mim[tid];
  CC[l * 32 + e * 2 + 0] = cp * mr - sp * mi;
  CC[l * 32 + e * 2 + 1] = cp * mi + sp * mr;
}

// 16 async b128 copies sharing one (global, LDS) base pair; per the ISA the
// instruction's INST_OFFSET is added to BOTH the global and the LDS address.
#define ALOAD16(ldsoff, gptr)                                               \
  asm volatile(                                                             \
      "global_load_async_to_lds_b128 %0, %1, off offset:0\n\t"              \
      "global_load_async_to_lds_b128 %0, %1, off offset:16\n\t"             \
      "global_load_async_to_lds_b128 %0, %1, off offset:32\n\t"             \
      "global_load_async_to_lds_b128 %0, %1, off offset:48\n\t"             \
      "global_load_async_to_lds_b128 %0, %1, off offset:64\n\t"             \
      "global_load_async_to_lds_b128 %0, %1, off offset:80\n\t"             \
      "global_load_async_to_lds_b128 %0, %1, off offset:96\n\t"             \
      "global_load_async_to_lds_b128 %0, %1, off offset:112\n\t"            \
      "global_load_async_to_lds_b128 %0, %1, off offset:128\n\t"            \
      "global_load_async_to_lds_b128 %0, %1, off offset:144\n\t"            \
      "global_load_async_to_lds_b128 %0, %1, off offset:160\n\t"            \
      "global_load_async_to_lds_b128 %0, %1, off offset:176\n\t"            \
      "global_load_async_to_lds_b128 %0, %1, off offset:192\n\t"            \
      "global_load_async_to_lds_b128 %0, %1, off offset:208\n\t"            \
      "global_load_async_to_lds_b128 %0, %1, off offset:224\n\t"            \
      "global_load_async_to_lds_b128 %0, %1, off offset:240\n\t"            \
      :: "v"(ldsoff), "v"(gptr) : "memory")

__global__ __launch_bounds__(256)
void fft_unitary_main(const float* __restrict__ X,
                      const float* __restrict__ AopRe, const float* __restrict__ AopIm,
                      const float* __restrict__ BopRe, const float* __restrict__ BopIm,
                      const float* __restrict__ CC,
                      float* __restrict__ out) {
  extern __shared__ float smem[];
  float* lre = smem;                 // dynamic LDS starts at offset 0
  float* lim = smem + 16 * LDST;
  const int tid = threadIdx.x;
  const int n0 = blockIdx.x * 16;

  // ---- async global->LDS fill: 16 rows (re + im), 256B segment per thread ----
  {
    const int n = tid >> 4, seg = tid & 15;          // row, 64-float segment
    const float* grow = X + (size_t)(n0 + n) * (2 * DD) + seg * 64;
    const unsigned lre_off = (unsigned)((n * LDST + seg * 64) * 4);
    const unsigned lim_off = (unsigned)(((16 * LDST) + n * LDST + seg * 64) * 4);
    ALOAD16(lre_off, grow);                          // real part
    ALOAD16(lim_off, grow + DD);                     // imaginary part
    asm volatile("s_wait_asynccnt 0x0" ::: "memory");
  }
  __syncthreads();

  const int w = tid >> 5, lane = tid & 31;
  const int hi = lane >> 4, nl = lane & 15;

  // ---- stage A: 64 groups of 16 contiguous elems; 8 groups per wave ----
  for (int gi = 0; gi < 8; ++gi) {
    const int g = w + gi * 8;
    v8f dr = {0.f, 0.f, 0.f, 0.f, 0.f, 0.f, 0.f, 0.f};
    v8f di = {0.f, 0.f, 0.f, 0.f, 0.f, 0.f, 0.f, 0.f};
    const float2* are = (const float2*)AopRe + g * 4 * 32 + lane;
    const float2* aim = (const float2*)AopIm + g * 4 * 32 + lane;
#pragma unroll
    for (int kc = 0; kc < 4; ++kc) {
      float2 ar2 = are[kc * 32], ai2 = aim[kc * 32];
      const int col = g * 16 + kc * 4 + 2 * hi;
      float2 br2 = *(const float2*)&lre[nl * LDST + col];
      float2 bi2 = *(const float2*)&lim[nl * LDST + col];
      v2f ar  = { ar2.x,  ar2.y};
      v2f ai  = { ai2.x,  ai2.y};
      v2f nai = {-ai2.x, -ai2.y};
      v2f br  = { br2.x,  br2.y};
      v2f bi  = { bi2.x,  bi2.y};
      dr = __builtin_amdgcn_wmma_f32_16x16x4_f32(false, ar,  false, br, (short)0, dr, false, false);
      dr = __builtin_amdgcn_wmma_f32_16x16x4_f32(false, nai, false, bi, (short)0, dr, false, false);
      di = __builtin_amdgcn_wmma_f32_16x16x4_f32(false, ar,  false, bi, (short)0, di, false, false);
      di = __builtin_amdgcn_wmma_f32_16x16x4_f32(false, ai,  false, br, (short)0, di, false, false);
    }
#pragma unroll
    for (int v = 0; v < 8; ++v) {            // C/D layout: VGPR v -> M = v + 8*hi, N = lane&15
      const int m = v + 8 * hi;
      lre[nl * LDST + g * 16 + m] = dr[v];
      lim[nl * LDST + g * 16 + m] = di[v];
    }
  }
  __syncthreads();

  // ---- stage B: 64 groups at stride 16; 8 groups per wave ----
  for (int hIdx = 0; hIdx < 8; ++hIdx) {
    const int h = w + hIdx * 8;
    const int base = (h >> 4) * 256 + (h & 15);
    v8f dr = {0.f, 0.f, 0.f, 0.f, 0.f, 0.f, 0.f, 0.f};
    v8f di = {0.f, 0.f, 0.f, 0.f, 0.f, 0.f, 0.f, 0.f};
    const float2* are = (const float2*)BopRe + h * 4 * 32 + lane;
    const float2* aim = (const float2*)BopIm + h * 4 * 32 + lane;
#pragma unroll
    for (int kc = 0; kc < 4; ++kc) {
      float2 ar2 = are[kc * 32], ai2 = aim[kc * 32];
      const int k0 = kc * 4 + 2 * hi;
      v2f br = {lre[nl * LDST + base + k0 * 16], lre[nl * LDST + base + (k0 + 1) * 16]};
      v2f bi = {lim[nl * LDST + base + k0 * 16], lim[nl * LDST + base + (k0 + 1) * 16]};
      v2f ar  = { ar2.x,  ar2.y};
      v2f ai  = { ai2.x,  ai2.y};
      v2f nai = {-ai2.x, -ai2.y};
      dr = __builtin_amdgcn_wmma_f32_16x16x4_f32(false, ar,  false, br, (short)0, dr, false, false);
      dr = __builtin_amdgcn_wmma_f32_16x16x4_f32(false, nai, false, bi, (short)0, dr, false, false);
      di = __builtin_amdgcn_wmma_f32_16x16x4_f32(false, ar,  false, bi, (short)0, di, false, false);
      di = __builtin_amdgcn_wmma_f32_16x16x4_f32(false, ai,  false, br, (short)0, di, false, false);
    }
#pragma unroll
    for (int v = 0; v < 8; ++v) {
      const int m = v + 8 * hi;
      lre[nl * LDST + base + m * 16] = dr[v];
      lim[nl * LDST + base + m * 16] = di[v];
    }
  }
  __syncthreads();

  // ---- stage C: 4x4 complex matvec per (row, l=tid), psi folded in ----
  float cr[16], ci[16];
  {
    const float2* cm = (const float2*)CC + tid * 16;
#pragma unroll
    for (int e = 0; e < 16; ++e) { float2 t = cm[e]; cr[e] = t.x; ci[e] = t.y; }
  }
  for (int nn = 0; nn < 16; ++nn) {
    float xr[4], xi[4];
#pragma unroll
    for (int m = 0; m < 4; ++m) {
      xr[m] = lre[nn * LDST + m * 256 + tid];
      xi[m] = lim[nn * LDST + m * 256 + tid];
    }
    float* orow = out + (size_t)(n0 + nn) * (2 * DD);
#pragma unroll
    for (int mp = 0; mp < 4; ++mp) {
      float yr = 0.f, yi = 0.f;
#pragma unroll
      for (int m = 0; m < 4; ++m) {
        yr = fmaf(cr[mp * 4 + m], xr[m], fmaf(-ci[mp * 4 + m], xi[m], yr));
        yi = fmaf(cr[mp * 4 + m], xi[m], fmaf( ci[mp * 4 + m], xr[m], yi));
      }
      orow[mp * 256 + tid]      = yr;   // Yr
      orow[DD + mp * 256 + tid] = yi;   // Yi
    }
  }
}

extern "C" void kernel_launch(void* const* d_in, const int* in_sizes, int n_in,
                              void* d_out, int out_size, void* d_ws, size_t ws_size,
                              hipStream_t stream) {
  const float* X     = (const float*)d_in[0];   // N x 2D
  const float* theta = (const float*)d_in[1];   // P x D/2
  const float* phi   = (const float*)d_in[2];   // P x D/2
  const float* psi   = (const float*)d_in[3];   // D
  float* out = (float*)d_out;

  // workspace layout (floats): A 64x4x32x2 re/im, B same, CC 256x4x4x2
  float* wsf   = (float*)d_ws;
  float* AopRe = wsf;
  float* AopIm = wsf + 16384;
  float* BopRe = wsf + 32768;
  float* BopIm = wsf + 49152;
  float* CCm   = wsf + 65536;   // 8192 floats; total 288 KB

  build_mats16<<<64, 256, 0, stream>>>(theta, phi, AopRe, AopIm, 0, 0);
  build_mats16<<<64, 256, 0, stream>>>(theta, phi, BopRe, BopIm, 4, 1);
  build_mats4 <<<16, 256, 0, stream>>>(theta, phi, psi, CCm);

  const size_t lds_bytes = (size_t)2 * 16 * LDST * sizeof(float); // 131584 B
  fft_unitary_main<<<NN / 16, 256, lds_bytes, stream>>>(X, AopRe, AopIm, BopRe, BopIm, CCm, out);
}